// Generator_42984032698976
// MI455X (gfx1250) — compile-verified
//
#include <hip/hip_runtime.h>

// ---- problem constants (match reference) ----
#define N_LINEARS        16
#define Z_DIM            128
#define BATCH            512
#define FEATS_PER_LINEAR 12288      // CHANNELS * OUTPUT_DIM^2 = 3*64*64
#define MTILE            16
#define NTILE            16
#define NT_PER_BLOCK     8          // 8 n-tiles -> 128 features per block
#define N_BLOCKS_N       (FEATS_PER_LINEAR / (NTILE * NT_PER_BLOCK))  // 96
#define MAX_MTILES       ((BATCH + MTILE - 1) / MTILE)                // 32

typedef __attribute__((ext_vector_type(2))) float v2f;
typedef __attribute__((ext_vector_type(8))) float v8f;

// -------------------------------------------------------------------------
// Kernel 1: per-sample bucket hash + build per-bucket sample lists.
// One block of BATCH threads; counts zeroed here (no cross-call state).
// -------------------------------------------------------------------------
__global__ void __launch_bounds__(BATCH)
bucket_kernel(const float* __restrict__ z, int* __restrict__ counts,
              int* __restrict__ idx) {
    const int b = threadIdx.x;
    if (b < N_LINEARS) counts[b] = 0;
    __syncthreads();

    const float* row = z + b * Z_DIM;
    float s = 0.0f;
#pragma unroll 8
    for (int k = 0; k < Z_DIM; ++k) s += row[k];
    s = fabsf(s) * 1000.0f;
    const int c = ((int)floorf(s)) & (N_LINEARS - 1);  // s >= 0, mod 16

    const int pos = atomicAdd(&counts[c], 1);
    idx[c * BATCH + pos] = b;
}

// -------------------------------------------------------------------------
// Kernel 2: grouped GEMM, one wave (32 lanes) per block.
//   A (16 x 4 f32): z rows of this bucket's samples
//   B (4 x 16 f32): W^T tile (W is row-major (out_features, Z_DIM))
//   D = A*B + C accumulated over K=128 via 32 x V_WMMA_F32_16X16X4_F32
// -------------------------------------------------------------------------
__global__ void __launch_bounds__(32)
grouped_gemm_kernel(const float* __restrict__ z,
                    const float* __restrict__ W,
                    const float* __restrict__ bias,
                    const int*   __restrict__ counts,
                    const int*   __restrict__ idx,
                    float*       __restrict__ out) {
    const int g  = blockIdx.z;   // bucket (linear id)
    const int mt = blockIdx.y;   // m-tile within bucket
    const int nb = blockIdx.x;   // n-block (128 features)

    const int cnt = counts[g];
    if (mt * MTILE >= cnt) return;           // uniform early-out

    const int lane = threadIdx.x;            // 0..31, wave32
    const int half = lane >> 4;              // 0 | 1
    const int l16  = lane & 15;

    // A-fragment source row: sample for M = l16 (padded rows borrow sample 0;
    // their results are never stored, so EXEC stays all-1s for WMMA).
    const int mrowA = mt * MTILE + l16;
    const int sampA = (mrowA < cnt) ? idx[g * BATCH + mrowA] : idx[g * BATCH];
    const float* zrow = z + (size_t)sampA * Z_DIM;

    const int gbase = g * FEATS_PER_LINEAR;            // bucket's first feature
    const int nbase = nb * (NTILE * NT_PER_BLOCK);     // within-bucket base

    // B-fragment source rows: W[(gbase + nbase + t*16 + l16)][k]
    const float* wrow0 = W + (size_t)(gbase + nbase + l16) * Z_DIM;

    v8f acc[NT_PER_BLOCK];
#pragma unroll
    for (int t = 0; t < NT_PER_BLOCK; ++t) acc[t] = (v8f){};

    const int koff = half * 2;   // lanes 0-15: K={0,1}; lanes 16-31: K={2,3}

    for (int kk = 0; kk < Z_DIM; kk += 4) {
        // A fragment: 16x4 tile, one float2 per lane (K-contiguous)
        v2f a = *(const v2f*)(zrow + kk + koff);
#pragma unroll
        for (int t = 0; t < NT_PER_BLOCK; ++t) {
            // B fragment: 4x16 tile of W^T, one float2 per lane
            v2f bfrag = *(const v2f*)(wrow0 + (size_t)t * NTILE * Z_DIM + kk + koff);
            acc[t] = __builtin_amdgcn_wmma_f32_16x16x4_f32(
                false, a, false, bfrag, (short)0, acc[t], false, false);
        }
    }

    // Epilogue: C/D layout -> lane covers M = v + half*8 (v=0..7), N = l16.
    int samp[8];
    bool mv[8];
#pragma unroll
    for (int v = 0; v < 8; ++v) {
        const int mrow = mt * MTILE + v + half * 8;
        mv[v]   = (mrow < cnt);
        samp[v] = mv[v] ? idx[g * BATCH + mrow] : 0;
    }

#pragma unroll
    for (int t = 0; t < NT_PER_BLOCK; ++t) {
        const int nfeat = nbase + t * NTILE + l16;     // within-bucket feature
        const float bv = bias[gbase + nfeat];
#pragma unroll
        for (int v = 0; v < 8; ++v) {
            if (mv[v]) {
                out[(size_t)samp[v] * FEATS_PER_LINEAR + nfeat] = acc[t][v] + bv;
            }
        }
    }
}

// -------------------------------------------------------------------------
// Host-side launcher (graph-capture safe: only kernel launches on `stream`).
// d_ws layout: int counts[16]; int idx[16][512]   (32832 B)
// -------------------------------------------------------------------------
extern "C" void kernel_launch(void* const* d_in, const int* in_sizes, int n_in,
                              void* d_out, int out_size, void* d_ws, size_t ws_size,
                              hipStream_t stream) {
    const float* z    = (const float*)d_in[0];
    const float* W    = (const float*)d_in[1];
    const float* bias = (const float*)d_in[2];
    float* out = (float*)d_out;

    int* counts = (int*)d_ws;
    int* idx    = counts + N_LINEARS;

    bucket_kernel<<<1, BATCH, 0, stream>>>(z, counts, idx);

    dim3 grid(N_BLOCKS_N, MAX_MTILES, N_LINEARS);
    grouped_gemm_kernel<<<grid, 32, 0, stream>>>(z, W, bias, counts, idx, out);
}